// FLAME_76828374991166
// MI455X (gfx1250) — compile-verified
//
#include <hip/hip_runtime.h>

typedef __attribute__((ext_vector_type(2))) float v2f;
typedef __attribute__((ext_vector_type(8))) float v8f;

#define B_SZ 1024
#define NV   5023
#define NJ   5
#define NVT  ((NV + 15) / 16)          // 314 vertex tiles
// workspace layout (floats):
//   A rows   [B][5][3][4]                      -> 61440 floats
//   posefeat [B][36]                           -> 36864 floats
//   pdT      [NVT][3][9][32 lanes][2]          -> 542592 floats (WMMA-B order)
#define WS_A_FLOATS   (B_SZ * NJ * 12)
#define WS_PF_FLOATS  (B_SZ * 36)
#define WS_PDT_FLOATS (NVT * 3 * 9 * 64)

// ---------------- kernel 1: per-batch pose prep (tiny) ----------------

__device__ __forceinline__ void rodrigues3(const float* rv, float R[3][3]) {
    float vx = rv[0] + 1e-8f, vy = rv[1] + 1e-8f, vz = rv[2] + 1e-8f;
    float angle = sqrtf(vx * vx + vy * vy + vz * vz);
    float inv = 1.0f / angle;
    float rx = rv[0] * inv, ry = rv[1] * inv, rz = rv[2] * inv;
    float c = cosf(angle), s = sinf(angle), t = 1.0f - c;
    R[0][0] = 1.0f - t * (ry * ry + rz * rz);
    R[0][1] = -s * rz + t * rx * ry;
    R[0][2] =  s * ry + t * rx * rz;
    R[1][0] =  s * rz + t * rx * ry;
    R[1][1] = 1.0f - t * (rx * rx + rz * rz);
    R[1][2] = -s * rx + t * ry * rz;
    R[2][0] = -s * ry + t * rx * rz;
    R[2][1] =  s * rx + t * ry * rz;
    R[2][2] = 1.0f - t * (rx * rx + ry * ry);
}

__device__ __forceinline__ void compose(float Ro[3][3], float* to,
                                        const float Ra[3][3], const float* ta,
                                        const float Rb[3][3], const float* tb) {
#pragma unroll
    for (int i = 0; i < 3; ++i) {
#pragma unroll
        for (int j = 0; j < 3; ++j)
            Ro[i][j] = Ra[i][0] * Rb[0][j] + Ra[i][1] * Rb[1][j] + Ra[i][2] * Rb[2][j];
        to[i] = Ra[i][0] * tb[0] + Ra[i][1] * tb[1] + Ra[i][2] * tb[2] + ta[i];
    }
}

__global__ void flame_pose_prep(const float* __restrict__ g6,
                                const float* __restrict__ neck,
                                const float* __restrict__ jaw,
                                const float* __restrict__ eye,
                                const float* __restrict__ Jr,
                                float* __restrict__ wsA,
                                float* __restrict__ wsPF) {
    int b = blockIdx.x * blockDim.x + threadIdx.x;
    if (b >= B_SZ) return;

    float R[5][3][3];
    {   // rot6d -> matrix (rows b1,b2,b3)
        const float* g = g6 + b * 6;
        float a1x = g[0], a1y = g[1], a1z = g[2];
        float a2x = g[3], a2y = g[4], a2z = g[5];
        float n1 = 1.0f / sqrtf(a1x * a1x + a1y * a1y + a1z * a1z);
        float b1x = a1x * n1, b1y = a1y * n1, b1z = a1z * n1;
        float d = b1x * a2x + b1y * a2y + b1z * a2z;
        float b2x = a2x - d * b1x, b2y = a2y - d * b1y, b2z = a2z - d * b1z;
        float n2 = 1.0f / sqrtf(b2x * b2x + b2y * b2y + b2z * b2z);
        b2x *= n2; b2y *= n2; b2z *= n2;
        float b3x = b1y * b2z - b1z * b2y;
        float b3y = b1z * b2x - b1x * b2z;
        float b3z = b1x * b2y - b1y * b2x;
        R[0][0][0] = b1x; R[0][0][1] = b1y; R[0][0][2] = b1z;
        R[0][1][0] = b2x; R[0][1][1] = b2y; R[0][1][2] = b2z;
        R[0][2][0] = b3x; R[0][2][1] = b3y; R[0][2][2] = b3z;
    }
    rodrigues3(neck + b * 3, R[1]);
    rodrigues3(jaw + b * 3, R[2]);
    rodrigues3(eye + b * 6, R[3]);
    rodrigues3(eye + b * 6 + 3, R[4]);

    float J[5][3];
#pragma unroll
    for (int j = 0; j < 5; ++j)
#pragma unroll
        for (int k = 0; k < 3; ++k) J[j][k] = Jr[(b * 5 + j) * 3 + k];

    float rel[5][3];
#pragma unroll
    for (int k = 0; k < 3; ++k) {
        rel[0][k] = J[0][k];
        rel[1][k] = J[1][k] - J[0][k];
        rel[2][k] = J[2][k] - J[1][k];
        rel[3][k] = J[3][k] - J[1][k];
        rel[4][k] = J[4][k] - J[1][k];
    }

    float Rc[5][3][3], tc[5][3];
#pragma unroll
    for (int i = 0; i < 3; ++i) {
        tc[0][i] = rel[0][i];
#pragma unroll
        for (int j = 0; j < 3; ++j) Rc[0][i][j] = R[0][i][j];
    }
    compose(Rc[1], tc[1], Rc[0], tc[0], R[1], rel[1]);
    compose(Rc[2], tc[2], Rc[1], tc[1], R[2], rel[2]);
    compose(Rc[3], tc[3], Rc[1], tc[1], R[3], rel[3]);
    compose(Rc[4], tc[4], Rc[1], tc[1], R[4], rel[4]);

    // A rows: [b][j][h][0..2] = Rc, [h][3] = tc - Rc*J  (top 3 rows only)
    float* Ab = wsA + b * 60;
#pragma unroll
    for (int j = 0; j < 5; ++j)
#pragma unroll
        for (int h = 0; h < 3; ++h) {
            float tcomp = tc[j][h] - (Rc[j][h][0] * J[j][0] + Rc[j][h][1] * J[j][1] +
                                      Rc[j][h][2] * J[j][2]);
            Ab[j * 12 + h * 4 + 0] = Rc[j][h][0];
            Ab[j * 12 + h * 4 + 1] = Rc[j][h][1];
            Ab[j * 12 + h * 4 + 2] = Rc[j][h][2];
            Ab[j * 12 + h * 4 + 3] = tcomp;
        }

    // pose_feat = (rot_mats[1:5] - I).flatten
    float* pf = wsPF + b * 36;
#pragma unroll
    for (int j = 1; j < 5; ++j)
#pragma unroll
        for (int h = 0; h < 3; ++h)
#pragma unroll
            for (int w = 0; w < 3; ++w)
                pf[(j - 1) * 9 + h * 3 + w] = R[j][h][w] - ((h == w) ? 1.0f : 0.0f);
}

// ------- kernel 1b: transpose posedirs into WMMA-B fragment order -------
// pdT[((vt*3 + c)*9 + s)*64 + lane*2 + i] = posedirs[v][p][c]
//   v = vt*16 + (lane&15), p = 4*s + i + 2*(lane>>4)
// -> each lane of the main kernel does ONE coalesced global_load_b64 per
//    (c,s) that lands directly in an even-aligned WMMA operand pair.

__global__ void flame_pdt_transpose(const float* __restrict__ posedirs,
                                    float* __restrict__ pdT) {
    int tid = blockIdx.x * blockDim.x + threadIdx.x;   // one thread = one lane slot
    if (tid >= NVT * 3 * 9 * 32) return;
    int lane = tid & 31;
    int rest = tid >> 5;
    int s = rest % 9;
    int rest2 = rest / 9;
    int c = rest2 % 3;
    int vt = rest2 / 3;

    int v = vt * 16 + (lane & 15);
    if (v >= NV) v = NV - 1;               // clamped duplicate; never stored to out
    int p0 = 4 * s + 2 * (lane >> 4);
    pdT[(size_t)tid * 2 + 0] = posedirs[((size_t)v * 36 + p0 + 0) * 3 + c];
    pdT[(size_t)tid * 2 + 1] = posedirs[((size_t)v * 36 + p0 + 1) * 3 + c];
}

// ---------------- kernel 2: fused WMMA pose-blend + LBS skinning ----------------

template <bool USE_PDT>
__global__ __launch_bounds__(256) void flame_lbs_wmma(
    const float* __restrict__ vshaped,   // (B,V,3)
    const float* __restrict__ lbs,       // (V,5)
    const float* __restrict__ posedirs,  // (V,36,3)
    const float* __restrict__ pdT,       // WMMA-B order (if USE_PDT)
    const float* __restrict__ wsA,       // (B,5,3,4)
    const float* __restrict__ wsPF,      // (B,36)
    float* __restrict__ out)             // (B,V,3)
{
    __shared__ __align__(16) float sPF[16 * 36];  // pose_feat tile
    __shared__ __align__(16) float sA[16 * 60];   // A rows tile [lb][j][h][w]

    const int lane = threadIdx.x & 31;
    const int wave = threadIdx.x >> 5;
    const int b0 = blockIdx.y * 16;

    // cooperative stage of per-batch data (reused by all 8 waves)
    for (int i = threadIdx.x; i < 16 * 36; i += 256) sPF[i] = wsPF[b0 * 36 + i];
    for (int i = threadIdx.x; i < 16 * 60; i += 256) sA[i]  = wsA[b0 * 60 + i];
    __syncthreads();

    const int vt = blockIdx.x * 8 + wave;
    const int v0 = vt * 16;
    if (v0 >= NV) return;                  // whole-wave uniform exit; EXEC stays full

    const int half = lane >> 4;            // 0: lanes 0-15, 1: lanes 16-31
    const int m = lane & 15;
    const int v = v0 + m;
    const int vc = (v < NV) ? v : (NV - 1);  // clamp loads; stores guarded later

    // A-matrix fragments (pose_feat, M=batch=m, K = 4s + i + 2*half)
    v2f af[9];
#pragma unroll
    for (int s = 0; s < 9; ++s) {
        af[s][0] = sPF[m * 36 + 4 * s + 2 * half + 0];
        af[s][1] = sPF[m * 36 + 4 * s + 2 * half + 1];
    }

    // skinning weights for this lane's vertex
    float w[5];
#pragma unroll
    for (int j = 0; j < 5; ++j) w[j] = lbs[vc * 5 + j];

    // v' = v_shaped + posedirs . pose_feat, per component, via chained f32 WMMA.
    // C is initialized with v_shaped (non-temporal: stream-once data).
    v8f accs[3];
#pragma unroll
    for (int c = 0; c < 3; ++c) {
        v8f acc;
        const float* vsb = vshaped + ((size_t)(b0 + half * 8) * NV + vc) * 3 + c;
#pragma unroll
        for (int r = 0; r < 8; ++r)
            acc[r] = __builtin_nontemporal_load(vsb + (size_t)r * NV * 3);

        const float* pb  = posedirs + ((size_t)vc * 36 + 2 * half) * 3 + c;
        const float* cb  = pdT + ((size_t)(vt * 3 + c) * 9) * 64 + lane * 2;
#pragma unroll
        for (int s = 0; s < 9; ++s) {
            v2f bf;
            if constexpr (USE_PDT) {
                bf = *(const v2f*)(cb + s * 64);        // coalesced b64, frag order
            } else {
                bf[0] = pb[(4 * s + 0) * 3];
                bf[1] = pb[(4 * s + 1) * 3];
            }
            acc = __builtin_amdgcn_wmma_f32_16x16x4_f32(
                false, af[s], false, bf, (short)0, acc, false, false);
        }
        accs[c] = acc;
    }

    // out = sum_j w_j * (A_j . [v';1]) ; C layout: reg r -> batch b0 + r + 8*half
#pragma unroll
    for (int r = 0; r < 8; ++r) {
        const int lb = half * 8 + r;
        const float x = accs[0][r], y = accs[1][r], z = accs[2][r];
        float o0 = 0.f, o1 = 0.f, o2 = 0.f;
#pragma unroll
        for (int j = 0; j < 5; ++j) {
            const float4* Aj = (const float4*)&sA[lb * 60 + j * 12];  // LDS broadcast
            float4 r0 = Aj[0], r1 = Aj[1], r2 = Aj[2];
            o0 += w[j] * (r0.x * x + r0.y * y + r0.z * z + r0.w);
            o1 += w[j] * (r1.x * x + r1.y * y + r1.z * z + r1.w);
            o2 += w[j] * (r2.x * x + r2.y * y + r2.z * z + r2.w);
        }
        if (v < NV) {
            size_t o = ((size_t)(b0 + lb) * NV + v) * 3;
            __builtin_nontemporal_store(o0, &out[o + 0]);
            __builtin_nontemporal_store(o1, &out[o + 1]);
            __builtin_nontemporal_store(o2, &out[o + 2]);
        }
    }
}

// ---------------- launch ----------------

extern "C" void kernel_launch(void* const* d_in, const int* in_sizes, int n_in,
                              void* d_out, int out_size, void* d_ws, size_t ws_size,
                              hipStream_t stream) {
    (void)in_sizes; (void)n_in; (void)out_size;
    const float* vshaped = (const float*)d_in[0];
    const float* g6      = (const float*)d_in[1];
    const float* neck    = (const float*)d_in[2];
    const float* jaw     = (const float*)d_in[3];
    const float* eye     = (const float*)d_in[4];
    const float* Jr      = (const float*)d_in[5];
    const float* lbsw    = (const float*)d_in[6];
    const float* posed   = (const float*)d_in[7];
    // d_in[8] = parents (int64); kinematic structure [0,0,1,1,1] is hardcoded.

    float* wsA  = (float*)d_ws;
    float* wsPF = wsA + WS_A_FLOATS;
    float* pdT  = wsPF + WS_PF_FLOATS;
    float* out  = (float*)d_out;

    flame_pose_prep<<<dim3((B_SZ + 255) / 256), dim3(256), 0, stream>>>(
        g6, neck, jaw, eye, Jr, wsA, wsPF);

    const size_t needed =
        (size_t)(WS_A_FLOATS + WS_PF_FLOATS + WS_PDT_FLOATS) * sizeof(float);
    dim3 grid((NVT + 7) / 8, B_SZ / 16);   // 40 x 64 blocks, 8 waves/block

    if (ws_size >= needed) {
        int nslots = NVT * 3 * 9 * 32;
        flame_pdt_transpose<<<dim3((nslots + 255) / 256), dim3(256), 0, stream>>>(
            posed, pdT);
        flame_lbs_wmma<true><<<grid, dim3(256), 0, stream>>>(
            vshaped, lbsw, posed, pdT, wsA, wsPF, out);
    } else {
        flame_lbs_wmma<false><<<grid, dim3(256), 0, stream>>>(
            vshaped, lbsw, posed, /*pdT=*/wsA, wsA, wsPF, out);
    }
}